// Encoder_16432544874989
// MI455X (gfx1250) — compile-verified
//
#include <hip/hip_runtime.h>
#include <cstdint>
#include <cstddef>

typedef __attribute__((ext_vector_type(2))) float v2f;
typedef __attribute__((ext_vector_type(8))) float v8f;

#define NEG_SLOPE 0.01f

// ---------------------------------------------------------------- utilities

__global__ void zero_f32_v4(float4* __restrict__ p, long long n4) {
  long long i = (long long)blockIdx.x * blockDim.x + threadIdx.x;
  if (i < n4) p[i] = make_float4(0.f, 0.f, 0.f, 0.f);
}

__global__ void init_deg(float* __restrict__ deg, int n) {
  int i = blockIdx.x * blockDim.x + threadIdx.x;
  if (i < n) deg[i] = 1.0f;  // self-loop contributes 1 to every node's degree
}

__global__ void deg_accum(const int* __restrict__ dst, float* __restrict__ deg, int E) {
  int e = blockIdx.x * blockDim.x + threadIdx.x;
  if (e < E) atomicAdd(&deg[dst[e]], 1.0f);
}

__global__ void compute_dinv(const float* __restrict__ deg, float* __restrict__ dinv, int n) {
  int i = blockIdx.x * blockDim.x + threadIdx.x;
  if (i < n) {
    float d = deg[i];
    dinv[i] = d > 0.0f ? rsqrtf(d) : 0.0f;
  }
}

__global__ void compute_norm(const int* __restrict__ src, const int* __restrict__ dst,
                             const float* __restrict__ dinv, float* __restrict__ norm, int E) {
  int e = blockIdx.x * blockDim.x + threadIdx.x;
  if (e < E) norm[e] = dinv[src[e]] * dinv[dst[e]];
}

// ---------------------------------------------------------------- WMMA GEMM
// Y[nrows x NOUT] = X[nrows x K] @ W[K x NOUT], fp32, exact precision.
// One block handles one 16-row M-tile; wave w handles N-tile w (16 cols).
// A-tile staged in LDS with padded stride (K+4): (K+4)%64==4, so the 16 rows
// of an A-fragment read hit 16 distinct bank groups (conflict-free).
template <int K, int NOUT>
__global__ void gcn_gemm_wmma(const float* __restrict__ X, const float* __restrict__ W,
                              float* __restrict__ Y) {
  constexpr int LDSROW = K + 4;
  __shared__ float At[16 * LDSROW];

  const int tid = threadIdx.x;
  const size_t mbase_row = (size_t)blockIdx.x * 16;

  // Cooperative load of the 16 x K x-tile into LDS (float4, coalesced).
  const float* __restrict__ xtile = X + mbase_row * K;
  for (int i = tid * 4; i < 16 * K; i += blockDim.x * 4) {
    const int r = i / K;
    const int c = i % K;            // K multiple of 4 -> c..c+3 in one row
    float4 v = *(const float4*)(xtile + (size_t)r * K + c);
    float* d = &At[r * LDSROW + c];
    d[0] = v.x; d[1] = v.y; d[2] = v.z; d[3] = v.w;
  }
  __syncthreads();

  const int wave = tid >> 5;                 // N-tile index (blockDim == NOUT/16 * 32)
  const int lane = tid & 31;
  const int lrow  = lane & 15;               // A: M row within tile / B,C: N column
  const int khalf = (lane >> 4) * 2;         // 0 (lanes 0-15) or 2 (lanes 16-31)
  const int n0    = wave * 16;

  v8f acc = {};
  for (int k0 = 0; k0 < K; k0 += 4) {
    // A fragment (ISA 32-bit A 16x4 layout): VGPR0 = K = k0+khalf, VGPR1 = +1
    const float* ap = &At[lrow * LDSROW + k0 + khalf];
    v2f a; a.x = ap[0]; a.y = ap[1];
    // B fragment: lane's column n0+lrow; lane-half selects K pair (mirrors A)
    const float* wp = W + (size_t)(k0 + khalf) * NOUT + n0 + lrow;
    v2f b; b.x = wp[0]; b.y = wp[NOUT];
    acc = __builtin_amdgcn_wmma_f32_16x16x4_f32(
        /*neg_a=*/false, a, /*neg_b=*/false, b,
        /*c_mod=*/(short)0, acc, /*reuse_a=*/false, /*reuse_b=*/false);
  }

  // C layout: VGPR v -> M = v + (lane<16 ? 0 : 8), N = lane&15
  float* yp = Y + mbase_row * NOUT + n0 + lrow;
  const int mofs = (lane < 16) ? 0 : 8;
#pragma unroll
  for (int v = 0; v < 8; ++v) {
    yp[(size_t)(mofs + v) * NOUT] = acc[v];
  }
}

// ---------------------------------------------------------------- scatter-add
// One thread per (edge, float4). F/4 threads per edge (32 or 16: 1 or 2 edges per wave ->
// index/norm loads are wave-coherent, gathers are b128 and coalesced across
// the wave). Atomic adds stay scalar f32 (hardware floor; all L2-resident).
template <int F>
__global__ void edge_scatter_v4(const int* __restrict__ src, const int* __restrict__ dst,
                                const float* __restrict__ norm, const float* __restrict__ h,
                                float* __restrict__ agg, int E) {
  constexpr int FV = F / 4;   // float4 chunks per edge (power of two)
  long long gid = (long long)blockIdx.x * blockDim.x + threadIdx.x;
  if (gid >= (long long)E * FV) return;
  const int e  = (int)(gid >> __builtin_ctz(FV));
  const int f4 = (int)(gid & (FV - 1));
  const int s = src[e];
  const int d = dst[e];
  const float w = norm[e];
  const float4 hv = *(const float4*)(h + (size_t)s * F + f4 * 4);
  float* ap = agg + (size_t)d * F + f4 * 4;
  atomicAdd(ap + 0, hv.x * w);
  atomicAdd(ap + 1, hv.y * w);
  atomicAdd(ap + 2, hv.z * w);
  atomicAdd(ap + 3, hv.w * w);
}

// layer-1 finalize: agg + dinv^2 * h (self-loop) + bias, leaky_relu -> out
template <int F>
__global__ void finalize_act_v4(const float* __restrict__ agg, const float* __restrict__ h,
                                const float* __restrict__ dinv, const float* __restrict__ bias,
                                float* __restrict__ out, int n) {
  constexpr int FV = F / 4;
  long long gid = (long long)blockIdx.x * blockDim.x + threadIdx.x;
  if (gid >= (long long)n * FV) return;
  const int i  = (int)(gid >> __builtin_ctz(FV));
  const int f4 = (int)(gid & (FV - 1));
  const float di2 = dinv[i] * dinv[i];
  const size_t base = gid * 4;
  const float4 av = *(const float4*)(agg + base);
  const float4 hv = *(const float4*)(h + base);
  const float4 bv = *(const float4*)(bias + f4 * 4);
  float4 r;
  r.x = av.x + di2 * hv.x + bv.x;
  r.y = av.y + di2 * hv.y + bv.y;
  r.z = av.z + di2 * hv.z + bv.z;
  r.w = av.w + di2 * hv.w + bv.w;
  r.x = (r.x >= 0.f) ? r.x : NEG_SLOPE * r.x;
  r.y = (r.y >= 0.f) ? r.y : NEG_SLOPE * r.y;
  r.z = (r.z >= 0.f) ? r.z : NEG_SLOPE * r.z;
  r.w = (r.w >= 0.f) ? r.w : NEG_SLOPE * r.w;
  *(float4*)(out + base) = r;
}

// layer-2 finalize (in place on d_out, no activation)
template <int F>
__global__ void finalize_lin_v4(float* __restrict__ agg, const float* __restrict__ h,
                                const float* __restrict__ dinv, const float* __restrict__ bias,
                                int n) {
  constexpr int FV = F / 4;
  long long gid = (long long)blockIdx.x * blockDim.x + threadIdx.x;
  if (gid >= (long long)n * FV) return;
  const int i  = (int)(gid >> __builtin_ctz(FV));
  const int f4 = (int)(gid & (FV - 1));
  const float di2 = dinv[i] * dinv[i];
  const size_t base = gid * 4;
  const float4 av = *(const float4*)(agg + base);
  const float4 hv = *(const float4*)(h + base);
  const float4 bv = *(const float4*)(bias + f4 * 4);
  float4 r;
  r.x = av.x + di2 * hv.x + bv.x;
  r.y = av.y + di2 * hv.y + bv.y;
  r.z = av.z + di2 * hv.z + bv.z;
  r.w = av.w + di2 * hv.w + bv.w;
  *(float4*)(agg + base) = r;
}

// ---------------------------------------------------------------- launcher

extern "C" void kernel_launch(void* const* d_in, const int* in_sizes, int n_in,
                              void* d_out, int out_size, void* d_ws, size_t ws_size,
                              hipStream_t stream) {
  const float* x   = (const float*)d_in[0];   // [N, 256]
  const int*   ei  = (const int*)d_in[1];     // [2, E] flat: src then dst
  const float* W1  = (const float*)d_in[2];   // [256, 128]
  const float* b1  = (const float*)d_in[3];   // [128]
  const float* W2  = (const float*)d_in[4];   // [128, 64]
  const float* b2  = (const float*)d_in[5];   // [64]
  float* out = (float*)d_out;                 // [N, 64]

  const int N = in_sizes[0] / 256;
  const int E = in_sizes[1] / 2;
  const int* src = ei;
  const int* dst = ei + E;

  // workspace layout (floats)
  float* ws   = (float*)d_ws;
  float* deg  = ws;                       // N
  float* dinv = deg + N;                  // N
  float* norm = dinv + N;                 // E
  float* h1   = norm + E;                 // N*128  (also reused as g [N*64])
  float* agg1 = h1 + (size_t)N * 128;     // N*128  (becomes h2 in place)
  float* h2 = agg1;
  float* g  = h1;

  const int T = 256;
  auto blk = [](long long n, int t) { return (unsigned)((n + t - 1) / t); };

  // 1) normalization coefficients
  init_deg<<<blk(N, T), T, 0, stream>>>(deg, N);
  deg_accum<<<blk(E, T), T, 0, stream>>>(dst, deg, E);
  compute_dinv<<<blk(N, T), T, 0, stream>>>(deg, dinv, N);
  compute_norm<<<blk(E, T), T, 0, stream>>>(src, dst, dinv, norm, E);

  // 2) layer 1 linear: h1 = x @ W1   (N divisible by 16: 100000 = 6250*16)
  gcn_gemm_wmma<256, 128><<<N / 16, 8 * 32, 0, stream>>>(x, W1, h1);

  // 3) layer 1 aggregation + bias + leaky_relu
  zero_f32_v4<<<blk((long long)N * 32, T), T, 0, stream>>>((float4*)agg1, (long long)N * 32);
  edge_scatter_v4<128><<<blk((long long)E * 32, T), T, 0, stream>>>(src, dst, norm, h1, agg1, E);
  finalize_act_v4<128><<<blk((long long)N * 32, T), T, 0, stream>>>(agg1, h1, dinv, b1, h2, N);

  // 4) layer 2 linear: g = h2 @ W2
  gcn_gemm_wmma<128, 64><<<N / 16, 4 * 32, 0, stream>>>(h2, W2, g);

  // 5) layer 2 aggregation + bias -> d_out
  zero_f32_v4<<<blk((long long)N * 16, T), T, 0, stream>>>((float4*)out, (long long)N * 16);
  edge_scatter_v4<64><<<blk((long long)E * 16, T), T, 0, stream>>>(src, dst, norm, g, out, E);
  finalize_lin_v4<64><<<blk((long long)N * 16, T), T, 0, stream>>>(out, g, dinv, b2, N);
}